// TensorBoard_4423816315107
// MI455X (gfx1250) — compile-verified
//
#include <hip/hip_runtime.h>
#include <stdint.h>

// Problem constants (from the reference)
#define BSQ 19
#define N2Q 361      // BS*BS board points == history length M
#define MQ  361
#define GQ  24       // groups per board
#define BLK 384      // 12 wave32 per workgroup; >= N2Q so one point per thread

#define AS1 __attribute__((address_space(1)))
#define AS3 __attribute__((address_space(3)))

typedef int v4i __attribute__((vector_size(16)));   // b128 payload type

#if __has_builtin(__builtin_amdgcn_s_wait_asynccnt)
#define WAIT_ASYNC0() __builtin_amdgcn_s_wait_asynccnt(0)
#else
#define WAIT_ASYNC0() asm volatile("s_wait_asynccnt 0" ::: "memory")
#endif

// Async global -> LDS copy of one dword (ASYNCcnt-tracked, bypasses VGPRs).
__device__ __forceinline__ void async_g2l_b32(const int* gsrc, int* ldst) {
#if __has_builtin(__builtin_amdgcn_global_load_async_to_lds_b32)
  __builtin_amdgcn_global_load_async_to_lds_b32(
      (AS1 int*)(uintptr_t)gsrc,
      (AS3 int*)(uint32_t)(uintptr_t)ldst,
      /*imm offset*/0, /*cpol*/0);
#else
  unsigned lds_off = (unsigned)(uintptr_t)ldst;
  unsigned long long ga = (unsigned long long)(uintptr_t)gsrc;
  asm volatile("global_load_async_to_lds_b32 %0, %1, off"
               :: "v"(lds_off), "v"(ga) : "memory");
#endif
}

// Async global -> LDS copy of 16 bytes (both sides must be 16B aligned).
__device__ __forceinline__ void async_g2l_b128(const int* gsrc, int* ldst) {
#if __has_builtin(__builtin_amdgcn_global_load_async_to_lds_b128)
  __builtin_amdgcn_global_load_async_to_lds_b128(
      (AS1 v4i*)(uintptr_t)gsrc,
      (AS3 v4i*)(uint32_t)(uintptr_t)ldst,
      /*imm offset*/0, /*cpol*/0);
#else
  async_g2l_b32(gsrc + 0, ldst + 0);
  async_g2l_b32(gsrc + 1, ldst + 1);
  async_g2l_b32(gsrc + 2, ldst + 2);
  async_g2l_b32(gsrc + 3, ldst + 3);
#endif
}

__global__ __launch_bounds__(BLK)
void superko_mask_kernel(const unsigned char* __restrict__ legal,   // (B,19,19) bool (1 byte)
                         const int*  __restrict__ cur_player,       // (B,)
                         const int*  __restrict__ cur_hash,         // (B,)
                         const int*  __restrict__ hash_hist,        // (B,M)
                         const int*  __restrict__ move_count,       // (B,)
                         const int*  __restrict__ Zpos,             // (N2,3)
                         const int*  __restrict__ members,          // (K,)
                         const int*  __restrict__ indptr,           // (R+1,)
                         const int*  __restrict__ gptr,             // (B+1,)
                         const int*  __restrict__ cap_local,        // (B,N2,4)
                         const float* __restrict__ dummy,           // (B,19,19)
                         float* __restrict__ out)                   // (B,19,19)
{
  // 368 = 361 rounded up so the padded 4-wide scan (up to L+3 <= 364) fits.
  __shared__ alignas(16) int lds_hist[368];   // this board's history row (+pad)
  __shared__ alignas(16) int lds_z[3 * N2Q];  // Zobrist table, Zpos[n*3 + c]
  __shared__ int lds_gx[GQ];                  // per-group capture XOR delta

  const int b   = blockIdx.x;
  const int tid = threadIdx.x;
  const size_t base = (size_t)b * N2Q;

  // Get the big cap_local stream for this board in flight (global_prefetch_b8)
  // while the LDS staging below is outstanding.
  {
    const char* capb = (const char*)(cap_local + base * 4);
    const int bytes = N2Q * 4 * 4;                // 5776 B
    for (int off = tid * 128; off < bytes; off += BLK * 128)
      __builtin_prefetch(capb + off, 0, 1);
  }

  // --- Phase 1: async-stage history row + Zobrist table into LDS -----------
  // History row stride is 1444 B (only 4B aligned) -> b32 per element.
  if (tid < MQ)
    async_g2l_b32(hash_hist + (size_t)b * MQ + tid, &lds_hist[tid]);
  // Zobrist table: 1083 dwords = 270 x b128 + 3 x b32 (both sides 16B aligned).
  if (tid < (3 * N2Q) / 4)                        // 270 threads
    async_g2l_b128(Zpos + tid * 4, &lds_z[tid * 4]);
  if (tid < (3 * N2Q) % 4)                        // 3 tail dwords
    async_g2l_b32(Zpos + 1080 + tid, &lds_z[1080 + tid]);
  WAIT_ASYNC0();          // each wave drains its own ASYNCcnt ...
  __syncthreads();        // ... then the barrier makes all LDS writes visible

  const int p   = cur_player[b];
  const int opp = 1 - p;
  const int ch  = cur_hash[b];
  int L = move_count[b];
  if (L > MQ) L = MQ;

  // --- Phase 2a: per-group XOR of removal deltas (24 groups, 6 stones) -----
  if (tid < GQ) {
    const int gg = gptr[b] + tid;
    const int s = indptr[gg];
    const int e = indptr[gg + 1];
    int x = 0;
    for (int j = s; j < e; ++j) {
      const int m = members[j];
      // D[m] = Z[opp+1][m] ^ Z[0][m]  (remove opponent stone -> empty)
      x ^= lds_z[m * 3 + (opp + 1)] ^ lds_z[m * 3];
    }
    lds_gx[tid] = x;
  }
  // --- Phase 2b: pad history so the scan can run in whole int4 chunks.
  // Duplicating a *valid* entry (hist[0], only when L>0) provably leaves the
  // membership result unchanged. Writes hit [L, L+2], reads index 0: no race.
  if (tid >= 64 && tid < 67 && L > 0)
    lds_hist[L + (tid - 64)] = lds_hist[0];
  __syncthreads();

  // --- Phase 3: per-point new hash + 4-wide history membership -------------
  const int nIter = (L + 3) >> 2;   // int4 chunks to scan (0 if L == 0)
  for (int n = tid; n < N2Q; n += BLK) {
    const size_t idx = base + n;
    float r = 0.0f;
    if (legal[idx]) {
      // place own stone: xor out empty, xor in player color
      const int place = lds_z[n * 3] ^ lds_z[n * 3 + 1 + p];
      // 4 neighbor capture indices: one 16B-aligned global_load_b128
      const int4 cc = *reinterpret_cast<const int4*>(cap_local + idx * 4);
      int cap = 0;
      if (cc.x >= 0) cap ^= lds_gx[cc.x];
      if (cc.y >= 0) cap ^= lds_gx[cc.y];
      if (cc.z >= 0) cap ^= lds_gx[cc.z];
      if (cc.w >= 0) cap ^= lds_gx[cc.w];
      const int nh = ch ^ place ^ cap;
      // membership of nh in hist[0:L] (== the reference's sort+searchsorted),
      // scanned 4 hashes per ds_load_b128
      bool found = false;
      const int4* h4 = reinterpret_cast<const int4*>(lds_hist);
      for (int j = 0; j < nIter; ++j) {
        const int4 h = h4[j];
        if (h.x == nh || h.y == nh || h.z == nh || h.w == nh) {
          found = true;
          break;
        }
      }
      if (!found) r = dummy[idx];
    }
    out[idx] = r;
  }
}

extern "C" void kernel_launch(void* const* d_in, const int* in_sizes, int n_in,
                              void* d_out, int out_size, void* d_ws, size_t ws_size,
                              hipStream_t stream) {
  const unsigned char* legal   = (const unsigned char*)d_in[0]; // bool -> 1B
  const int*  cur_player = (const int*)d_in[1];
  const int*  cur_hash   = (const int*)d_in[2];
  const int*  hash_hist  = (const int*)d_in[3];
  const int*  move_count = (const int*)d_in[4];
  const int*  Zpos       = (const int*)d_in[5];
  const int*  members    = (const int*)d_in[6];
  const int*  indptr     = (const int*)d_in[7];
  const int*  gptr       = (const int*)d_in[8];
  const int*  cap_local  = (const int*)d_in[9];
  const float* dummy     = (const float*)d_in[10];
  float* out = (float*)d_out;

  const int Bn = in_sizes[1];   // boards = len(current_player)
  superko_mask_kernel<<<dim3(Bn), dim3(BLK), 0, stream>>>(
      legal, cur_player, cur_hash, hash_hist, move_count, Zpos,
      members, indptr, gptr, cap_local, dummy, out);
}